// ConstantBaseLine_9852654977121
// MI455X (gfx1250) — compile-verified
//
#include <hip/hip_runtime.h>

// Forward-fill ("constant baseline") over rows: out[b][n] = A[b][m] where m is
// the most recent index <= n with W[b][m]==0 (index 0 always counts).
// B=4096 rows, S=8192 cols, f32/i32. Memory-bound: ~402 MB streamed once
// => ~17us floor at 23.3 TB/s. One wave32 per row, 256-element chunks
// (8 elems/lane, 2x b128 per array), ballot + ds_bpermute + readlane scan.

#define ROWS 4096
#define COLS 8192
#define WAVES_PER_BLOCK 8
#define ELEMS 8                       // elements per lane per chunk
#define CHUNK (32 * ELEMS)            // 256
#define NCHUNK (COLS / CHUNK)         // 32

__global__ __launch_bounds__(WAVES_PER_BLOCK * 32)
void ffill_rows_kernel(const float* __restrict__ A,
                       const int*   __restrict__ W,
                       float*       __restrict__ O)
{
    const int lane = threadIdx.x & 31;
    const int wave = threadIdx.x >> 5;
    const int row  = blockIdx.x * WAVES_PER_BLOCK + wave;

    const long long rowOff = (long long)row * COLS;
    const float4* __restrict__ A4 = (const float4*)(A + rowOff);
    const int4*   __restrict__ W4 = (const int4*)(W + rowOff);
    float4*       __restrict__ O4 = (float4*)(O + rowOff);

    // "Index 0 always counts": seed the carry with A[row][0].
    float carry = A[rowOff];

    // float4-unit index of this lane's first quad within chunk c: c*64 + 2*lane.
    const int li = 2 * lane;

    // Software pipeline: current chunk in (a0,a1,w0,w1), next chunk in (*n).
    float4 a0 = A4[li],     a1 = A4[li + 1];
    int4   w0 = W4[li],     w1 = W4[li + 1];

    for (int c = 0; c < NCHUNK; ++c) {
        float4 a0n = a0, a1n = a1;
        int4   w0n = w0, w1n = w1;
        if (c + 1 < NCHUNK) {                 // wave-uniform branch
            const int nb = (c + 1) * 64 + li;
            a0n = A4[nb];  a1n = A4[nb + 1];
            w0n = W4[nb];  w1n = W4[nb + 1];
        }
        if (c + 2 < NCHUNK) {                 // deepen the pipeline via L2 prefetch
            const int pb = (c + 2) * 64 + li;
            __builtin_prefetch(&A4[pb], 0, 0);
            __builtin_prefetch(&W4[pb], 0, 0);
        }

        // Lane-local dry flags for the 8 owned elements.
        const bool h0 = (w0.x == 0), h1 = (w0.y == 0), h2 = (w0.z == 0), h3 = (w0.w == 0);
        const bool h4 = (w1.x == 0), h5 = (w1.y == 0), h6 = (w1.z == 0), h7 = (w1.w == 0);
        const bool has = h0 | h1 | h2 | h3 | h4 | h5 | h6 | h7;

        // Last dry value within this lane's 8 elements.
        float last = h0 ? a0.x : 0.0f;
        last       = h1 ? a0.y : last;
        last       = h2 ? a0.z : last;
        last       = h3 ? a0.w : last;
        last       = h4 ? a1.x : last;
        last       = h5 ? a1.y : last;
        last       = h6 ? a1.z : last;
        last       = h7 ? a1.w : last;

        // Wave-level: last dry value from any lane strictly below me.
        const unsigned mask  = __builtin_amdgcn_ballot_w32(has);
        const unsigned prior = mask & ((1u << lane) - 1u);
        const int src = 31 - __builtin_clz(prior | 1u);   // 0 when prior==0 (unused then)
        // Full-EXEC bpermute (all lanes active -> no disabled-source zeros).
        const float gathered = __int_as_float(
            __builtin_amdgcn_ds_bpermute(src << 2, __float_as_int(last)));
        const float ext = prior ? gathered : carry;

        // Serial 8-element fill starting from the external value.
        float r = ext;
        const float o0 = h0 ? a0.x : r; r = o0;
        const float o1 = h1 ? a0.y : r; r = o1;
        const float o2 = h2 ? a0.z : r; r = o2;
        const float o3 = h3 ? a0.w : r; r = o3;
        const float o4 = h4 ? a1.x : r; r = o4;
        const float o5 = h5 ? a1.y : r; r = o5;
        const float o6 = h6 ? a1.z : r; r = o6;
        const float o7 = h7 ? a1.w : r; r = o7;

        const int ob = c * 64 + li;
        O4[ob]     = make_float4(o0, o1, o2, o3);
        O4[ob + 1] = make_float4(o4, o5, o6, o7);

        // Carry = running value at the last element of the chunk (lane 31).
        carry = __int_as_float(__builtin_amdgcn_readlane(__float_as_int(r), 31));

        a0 = a0n; a1 = a1n;
        w0 = w0n; w1 = w1n;
    }
}

extern "C" void kernel_launch(void* const* d_in, const int* in_sizes, int n_in,
                              void* d_out, int out_size, void* d_ws, size_t ws_size,
                              hipStream_t stream) {
    const float* A = (const float*)d_in[0];
    const int*   W = (const int*)d_in[1];
    float*       O = (float*)d_out;

    dim3 block(WAVES_PER_BLOCK * 32);
    dim3 grid(ROWS / WAVES_PER_BLOCK);
    hipLaunchKernelGGL(ffill_rows_kernel, grid, block, 0, stream, A, W, O);
}